// Attn_25220047962768
// MI455X (gfx1250) — compile-verified
//
#include <hip/hip_runtime.h>
#include <hip/hip_bf16.h>

typedef __attribute__((ext_vector_type(2))) float v2f;
typedef __attribute__((ext_vector_type(8))) float v8f;

#define OUT_LEN 128
#define IN_LEN  1024
#define BATCH   4
#define HID     128

__device__ __forceinline__ float fast_tanh(float x) {
#if __has_builtin(__builtin_amdgcn_tanhf)
    return __builtin_amdgcn_tanhf(x);   // gfx1250 v_tanh_f32 (TRANS op)
#else
    return tanhf(x);
#endif
}

// ---------------------------------------------------------------------------
// Kernel 1: projections via V_WMMA_F32_16X16X4_F32.
//   dec_proj[r][k] = sum_h dec_flat[r][h] * attn_W[k][h]        + b[k]   (r = o*B+b, 512 rows)
//   enc_proj[r][k] = sum_h enc_flat[r][h] * attn_W[k][128+h]             (r = i*B+b, 4096 rows)
// One wave (32 threads) per 16x16 output tile; k-loop over H in steps of 4.
// Tiles 0..255 -> dec_proj (32x8 tiles), tiles 256..2303 -> enc_proj (256x8).
// ---------------------------------------------------------------------------
__global__ __launch_bounds__(32)
void proj_kernel(const float* __restrict__ dec,    // 512 x 128
                 const float* __restrict__ enc,    // 4096 x 128
                 const float* __restrict__ attn_W, // 128 x 256
                 const float* __restrict__ attn_b, // 128
                 float* __restrict__ dec_proj,     // 512 x 128
                 float* __restrict__ enc_proj)     // 4096 x 128
{
    int tile = blockIdx.x;
    const float* X;
    float* P;
    int rt, ct, wOff;
    bool addBias;
    if (tile < 256) {
        X = dec; P = dec_proj; rt = tile >> 3; ct = tile & 7; wOff = 0;   addBias = true;
    } else {
        int t = tile - 256;
        X = enc; P = enc_proj; rt = t >> 3;    ct = t & 7;    wOff = HID; addBias = false;
    }
    const int row0 = rt << 4;
    const int col0 = ct << 4;

    const int lane = threadIdx.x;      // 0..31, EXEC all ones
    const int hl   = lane >> 4;        // half: 0 or 1
    const int lid  = lane & 15;

    const float* xrow = X + (size_t)(row0 + lid) * HID;                  // A: row m=lid
    const float* wrow = attn_W + (size_t)(col0 + lid) * (2*HID) + wOff;  // B: col n=lid

    v8f c = {};
    #pragma unroll 4
    for (int h0 = 0; h0 < HID; h0 += 4) {
        const int h = h0 + 2 * hl;
        // 32-bit A 16x4 layout: lanes 0-15 hold K=0,1; lanes 16-31 hold K=2,3
        v2f a; a.x = xrow[h]; a.y = xrow[h + 1];
        // 32-bit B 4x16 layout: lane = N column; lanes 0-15 rows K=0,1; 16-31 rows K=2,3
        v2f b; b.x = wrow[h]; b.y = wrow[h + 1];
        c = __builtin_amdgcn_wmma_f32_16x16x4_f32(false, a, false, b, (short)0, c,
                                                  false, false);
    }

    const float bias = addBias ? attn_b[col0 + lid] : 0.0f;
    #pragma unroll
    for (int r = 0; r < 8; ++r) {
        const int m = row0 + r + 8 * hl;    // C/D layout: VGPR r -> M=r (lanes 0-15), M=r+8 (16-31)
        P[(size_t)m * HID + col0 + lid] = c[r] + bias;
    }
}

// ---------------------------------------------------------------------------
// Kernel 2: fused score + softmax (the FLOP hotspot: 67M tanh + FMA).
//   s[i] = sum_k tanh(dec_proj[o,b,k] + enc_proj[i,b,k]) * v[k]
//   weights[b,o,i] = softmax_i(s)
// One workgroup per (b,o) row: 256 threads, each owns 4 encoder positions.
// k-chunk is the OUTER loop so each dp/vv float4 (one b128 DS read each) is
// reused across 4 encoder streams -> 4x less LDS traffic, 4 VMEM streams in
// flight per thread.
// ---------------------------------------------------------------------------
__global__ __launch_bounds__(256)
void score_kernel(const float* __restrict__ dec_proj,  // [o*B+b][k]
                  const float* __restrict__ enc_proj,  // [i*B+b][k]
                  const float* __restrict__ v,         // [k]
                  float* __restrict__ weights)         // [b][o][i]
{
    __shared__ float4 dp[HID / 4];
    __shared__ float4 vv[HID / 4];
    __shared__ float  red[256];

    const int r   = blockIdx.x;       // r = o*B + b
    const int b   = r & (BATCH - 1);
    const int o   = r >> 2;
    const int tid = threadIdx.x;

    if (tid < HID / 4) {
        dp[tid] = reinterpret_cast<const float4*>(dec_proj + (size_t)r * HID)[tid];
        vv[tid] = reinterpret_cast<const float4*>(v)[tid];
    }
    __syncthreads();

    const float4* ep0 = reinterpret_cast<const float4*>(
        enc_proj + (size_t)((0 * 256 + tid) * BATCH + b) * HID);
    const float4* ep1 = reinterpret_cast<const float4*>(
        enc_proj + (size_t)((1 * 256 + tid) * BATCH + b) * HID);
    const float4* ep2 = reinterpret_cast<const float4*>(
        enc_proj + (size_t)((2 * 256 + tid) * BATCH + b) * HID);
    const float4* ep3 = reinterpret_cast<const float4*>(
        enc_proj + (size_t)((3 * 256 + tid) * BATCH + b) * HID);

    float s0 = 0.f, s1 = 0.f, s2 = 0.f, s3 = 0.f;
    #pragma unroll 4
    for (int k4 = 0; k4 < HID / 4; ++k4) {
        const float4 d = dp[k4];          // one ds_load_b128, shared by 4 streams
        const float4 w = vv[k4];
        const float4 e0 = ep0[k4];
        const float4 e1 = ep1[k4];
        const float4 e2 = ep2[k4];
        const float4 e3 = ep3[k4];
        s0 += fast_tanh(d.x + e0.x) * w.x + fast_tanh(d.y + e0.y) * w.y +
              fast_tanh(d.z + e0.z) * w.z + fast_tanh(d.w + e0.w) * w.w;
        s1 += fast_tanh(d.x + e1.x) * w.x + fast_tanh(d.y + e1.y) * w.y +
              fast_tanh(d.z + e1.z) * w.z + fast_tanh(d.w + e1.w) * w.w;
        s2 += fast_tanh(d.x + e2.x) * w.x + fast_tanh(d.y + e2.y) * w.y +
              fast_tanh(d.z + e2.z) * w.z + fast_tanh(d.w + e2.w) * w.w;
        s3 += fast_tanh(d.x + e3.x) * w.x + fast_tanh(d.y + e3.y) * w.y +
              fast_tanh(d.z + e3.z) * w.z + fast_tanh(d.w + e3.w) * w.w;
    }

    // block max
    float lm = fmaxf(fmaxf(s0, s1), fmaxf(s2, s3));
    red[tid] = lm;
    __syncthreads();
    for (int off = 128; off > 0; off >>= 1) {
        if (tid < off) red[tid] = fmaxf(red[tid], red[tid + off]);
        __syncthreads();
    }
    const float m = red[0];
    __syncthreads();

    const float e0v = __expf(s0 - m);
    const float e1v = __expf(s1 - m);
    const float e2v = __expf(s2 - m);
    const float e3v = __expf(s3 - m);
    red[tid] = e0v + e1v + e2v + e3v;
    __syncthreads();
    for (int off = 128; off > 0; off >>= 1) {
        if (tid < off) red[tid] += red[tid + off];
        __syncthreads();
    }
    const float inv = 1.0f / red[0];

    float* wrow = weights + ((size_t)b * OUT_LEN + o) * IN_LEN;
    wrow[0 * 256 + tid] = e0v * inv;
    wrow[1 * 256 + tid] = e1v * inv;
    wrow[2 * 256 + tid] = e2v * inv;
    wrow[3 * 256 + tid] = e3v * inv;
}

// ---------------------------------------------------------------------------
// Kernel 3: context = weights @ enc, per batch, via V_WMMA_F32_16X16X4_F32.
//   out[o][b][h] = sum_i weights[b][o][i] * enc[i][b][h]
// One wave per 16x16 (o,h) tile; K-loop over I=1024 in steps of 4 (256 WMMAs).
// ---------------------------------------------------------------------------
__global__ __launch_bounds__(32)
void context_kernel(const float* __restrict__ weights, // [b][o][i]
                    const float* __restrict__ enc,     // [i][b][h] (= encoder_outputs flat)
                    float* __restrict__ out)           // [o][b][h]
{
    const int t   = blockIdx.x;        // 0..255
    const int b   = t >> 6;
    const int rem = t & 63;
    const int row0 = (rem >> 3) << 4;  // o tile
    const int col0 = (rem & 7) << 4;   // h tile

    const int lane = threadIdx.x;
    const int hl   = lane >> 4;
    const int lid  = lane & 15;

    const float* arow = weights + ((size_t)b * OUT_LEN + row0 + lid) * IN_LEN;
    const float* bcol = enc + (size_t)b * HID + col0 + lid;

    v8f c = {};
    #pragma unroll 4
    for (int i0 = 0; i0 < IN_LEN; i0 += 4) {
        const int i = i0 + 2 * hl;
        v2f a; a.x = arow[i];
               a.y = arow[i + 1];
        v2f bb; bb.x = bcol[(size_t)i * (BATCH * HID)];
                bb.y = bcol[(size_t)(i + 1) * (BATCH * HID)];
        c = __builtin_amdgcn_wmma_f32_16x16x4_f32(false, a, false, bb, (short)0, c,
                                                  false, false);
    }

    #pragma unroll
    for (int r = 0; r < 8; ++r) {
        const int m = row0 + r + 8 * hl;   // o index
        out[(size_t)m * (BATCH * HID) + (size_t)b * HID + col0 + lid] = c[r];
    }
}

// ---------------------------------------------------------------------------
extern "C" void kernel_launch(void* const* d_in, const int* in_sizes, int n_in,
                              void* d_out, int out_size, void* d_ws, size_t ws_size,
                              hipStream_t stream) {
    const float* dec    = (const float*)d_in[0];  // (128, 4, 128)
    const float* enc    = (const float*)d_in[1];  // (1024, 4, 128)
    const float* attn_W = (const float*)d_in[2];  // (128, 256)
    const float* attn_b = (const float*)d_in[3];  // (128,)
    const float* v      = (const float*)d_in[4];  // (128,)
    float* out = (float*)d_out;                   // (128, 4, 128)

    float* dec_proj = (float*)d_ws;                                  // 512*128
    float* enc_proj = dec_proj + (size_t)OUT_LEN * BATCH * HID;      // 4096*128
    float* weights  = enc_proj + (size_t)IN_LEN * BATCH * HID;       // 4*128*1024

    // K1: 256 dec tiles + 2048 enc tiles, one wave each
    proj_kernel<<<2304, 32, 0, stream>>>(dec, enc, attn_W, attn_b, dec_proj, enc_proj);
    // K2: one workgroup per (b,o)
    score_kernel<<<OUT_LEN * BATCH, 256, 0, stream>>>(dec_proj, enc_proj, v, weights);
    // K3: 4 batches x 64 tiles, one wave each
    context_kernel<<<256, 32, 0, stream>>>(weights, enc, out);
}